// HyperbolicAttention_18296560681633
// MI455X (gfx1250) — compile-verified
//
#include <hip/hip_runtime.h>
#include <math.h>

typedef __attribute__((ext_vector_type(2))) float v2f;
typedef __attribute__((ext_vector_type(8))) float v8f;

#define EPSF   1e-8f
#define CLAMPV (1.0f - 1e-5f)

static constexpr int Bb = 2, Tt = 1024, Dd = 1024, Hh = 16, DK = 64;
static constexpr int NR = Bb * Tt; // 2048 rows

// ---------------------------------------------------------------- reductions
__device__ __forceinline__ float block_reduce_sum(float v, float* sm) {
  int t = threadIdx.x;
  sm[t] = v;
  __syncthreads();
  for (int s = blockDim.x >> 1; s > 0; s >>= 1) {
    if (t < s) sm[t] += sm[t + s];
    __syncthreads();
  }
  float r = sm[0];
  __syncthreads();
  return r;
}

// ------------------------------------------------- expmap0 of bias -> ball
__global__ void __launch_bounds__(256)
bias_ball_kernel(const float* __restrict__ b, float* __restrict__ pb,
                 float* __restrict__ meta) {
  __shared__ float sm[256];
  float s = 0.f;
  for (int j = threadIdx.x; j < Dd; j += 256) { float v = b[j]; s += v * v; }
  float n2 = block_reduce_sum(s, sm);
  float n = fmaxf(sqrtf(n2), EPSF);
  float t = tanhf(fminf(n, 15.0f));
  float factor = t / n;
  float xn = fmaxf(factor * sqrtf(n2), EPSF);
  if (xn > CLAMPV) factor *= CLAMPV / xn;
  for (int j = threadIdx.x; j < Dd; j += 256) pb[j] = factor * b[j];
  if (threadIdx.x == 0) meta[0] = factor * factor * n2; // |pb|^2
}

// ------------------------------------------------------------- logmap0(x)
__global__ void __launch_bounds__(256)
logmap_kernel(const float* __restrict__ x, float* __restrict__ u) {
  __shared__ float sm[256];
  const float* xr = x + (size_t)blockIdx.x * Dd;
  float s = 0.f;
  for (int j = threadIdx.x; j < Dd; j += 256) { float v = xr[j]; s += v * v; }
  float n2 = block_reduce_sum(s, sm);
  float n = fmaxf(sqrtf(n2), EPSF);
  float a = fminf(n, 1.0f - EPSF);
  float factor = atanhf(a) / n;
  float* ur = u + (size_t)blockIdx.x * Dd;
  for (int j = threadIdx.x; j < Dd; j += 256) ur[j] = factor * xr[j];
}

// --------------------------------------------- Y = A @ W^T (+bias), f32 WMMA
// A: M x K row-major.  W: N x K row-major (so B[k][n] = W[n][k]).
// Block = 8 waves; wave computes one 16x16 tile; block covers 64x32.
__global__ void __launch_bounds__(256)
gemm_wmma_kernel(const float* __restrict__ Am, const float* __restrict__ Wm,
                 const float* __restrict__ bias, float* __restrict__ Y,
                 int M, int N, int Kd) {
  const int w    = threadIdx.x >> 5;
  const int lane = threadIdx.x & 31;
  const int half = lane >> 4, l16 = lane & 15;
  const int r0 = blockIdx.y * 64 + (w & 3) * 16;
  const int c0 = blockIdx.x * 32 + (w >> 2) * 16;
  const float* arow = Am + (size_t)(r0 + l16) * Kd + half * 2;
  const float* brow = Wm + (size_t)(c0 + l16) * Kd + half * 2;
  v8f acc = {};
  for (int k0 = 0; k0 < Kd; k0 += 4) {
    v2f a = *(const v2f*)(arow + k0);
    v2f b = *(const v2f*)(brow + k0);
    acc = __builtin_amdgcn_wmma_f32_16x16x4_f32(false, a, false, b,
                                                (short)0, acc, false, false);
  }
  const int col = c0 + l16;
  float bv = bias ? bias[col] : 0.0f;
#pragma unroll
  for (int v = 0; v < 8; ++v) {
    int row = r0 + v + 8 * half;
    Y[(size_t)row * N + col] = acc[v] + bv;
  }
}

// ----------------- in-place: Y_row -> mobius_add(expmap0(Y_row), pb), c = 1
__global__ void __launch_bounds__(256)
hyp_epilogue_kernel(float* __restrict__ Y, const float* __restrict__ pb,
                    const float* __restrict__ meta) {
  __shared__ float sm[256];
  float* y = Y + (size_t)blockIdx.x * Dd;
  float svv = 0.f, svp = 0.f;
  for (int j = threadIdx.x; j < Dd; j += 256) {
    float v = y[j];
    svv += v * v;
    svp += v * pb[j];
  }
  float tot_vv = block_reduce_sum(svv, sm);
  float tot_vp = block_reduce_sum(svp, sm);
  // expmap0 of the row
  float y2 = meta[0];
  float n = fmaxf(sqrtf(tot_vv), EPSF);
  float t = tanhf(fminf(n, 15.0f));
  float factor = t / n;
  float xn = fmaxf(factor * sqrtf(tot_vv), EPSF);
  if (xn > CLAMPV) factor *= CLAMPV / xn;
  float x2 = factor * factor * tot_vv;
  float xy = factor * tot_vp;
  // mobius_add closed form:  (alpha*x + beta*pb)/den, clamped to ball
  float alpha = 1.0f + 2.0f * xy + y2;
  float beta  = 1.0f - x2;
  float den   = fmaxf(1.0f + 2.0f * xy + x2 * y2, EPSF);
  float num2  = alpha * alpha * x2 + 2.0f * alpha * beta * xy + beta * beta * y2;
  float rn = fmaxf(sqrtf(fmaxf(num2, 0.0f)) / den, EPSF);
  float cf = (rn > CLAMPV) ? (CLAMPV / rn) : 1.0f;
  float sA = cf * alpha * factor / den;  // multiplies original row value
  float sB = cf * beta / den;            // multiplies pb
  for (int j = threadIdx.x; j < Dd; j += 256) y[j] = sA * y[j] + sB * pb[j];
}

// ----------------------------------------- per-(b,t,h) squared norms over dk
__global__ void __launch_bounds__(64)
headnorm_kernel(const float* __restrict__ X, float* __restrict__ out) {
  __shared__ float sm[64];
  int flat = blockIdx.x;         // (b*Tt + t)*Hh + h
  int h = flat % Hh;
  int bt = flat / Hh;
  const float* p = X + (size_t)bt * Dd + h * DK;
  float v = p[threadIdx.x];
  sm[threadIdx.x] = v * v;
  __syncthreads();
  for (int st = 32; st > 0; st >>= 1) {
    if (threadIdx.x < st) sm[threadIdx.x] += sm[threadIdx.x + st];
    __syncthreads();
  }
  if (threadIdx.x == 0) {
    int b = bt / Tt, tt = bt % Tt;
    out[(size_t)(b * Hh + h) * Tt + tt] = sm[0];
  }
}

// -------------------- flash attention with Poincare-distance scores (c = 1)
// grid (Tt/128, Bb*Hh), block 256.  Wave handles 16 query rows.
__global__ void __launch_bounds__(256)
attn_kernel(const float* __restrict__ Q, const float* __restrict__ K,
            const float* __restrict__ V, const float* __restrict__ q2,
            const float* __restrict__ k2, float* __restrict__ O) {
  __shared__ float s_tile[8][256];
  __shared__ float s_scl[8][16];
  __shared__ float s_l[8][16];
  const int w    = threadIdx.x >> 5;
  const int lane = threadIdx.x & 31;
  const int half = lane >> 4, l16 = lane & 15;
  const int bh = blockIdx.y, b = bh / Hh, h = bh % Hh;
  const int qt = blockIdx.x * 128 + w * 16;
  const float* Qp = Q + (size_t)(b * Tt) * Dd + h * DK;
  const float* Kp = K + (size_t)(b * Tt) * Dd + h * DK;
  const float* Vp = V + (size_t)(b * Tt) * Dd + h * DK;
  const float* q2p = q2 + (size_t)bh * Tt;
  const float* k2p = k2 + (size_t)bh * Tt;

  // Preload Q A-fragments (16 K-steps of 4 over dk=64)
  v2f qa[16];
#pragma unroll
  for (int ks = 0; ks < 16; ++ks)
    qa[ks] = *(const v2f*)(Qp + (size_t)(qt + l16) * Dd + ks * 4 + half * 2);

  float q2v[8];
#pragma unroll
  for (int v = 0; v < 8; ++v) q2v[v] = q2p[qt + v + 8 * half];

  v8f acc[4] = {};
  float m_r = -3.0e38f, l_r = 0.0f;

  for (int kt = 0; kt < Tt / 16; ++kt) {
    const int jb = kt * 16;
    // qk tile = Q(16x64) @ K_tile^T(64x16) via 16 f32 WMMAs
    v8f s = {};
#pragma unroll
    for (int ks = 0; ks < 16; ++ks) {
      v2f bf = *(const v2f*)(Kp + (size_t)(jb + l16) * Dd + ks * 4 + half * 2);
      s = __builtin_amdgcn_wmma_f32_16x16x4_f32(false, qa[ks], false, bf,
                                                (short)0, s, false, false);
    }
    float k2j = k2p[jb + l16];
    // closed-form mobius distance -> score, write into LDS (row i, col j)
#pragma unroll
    for (int v = 0; v < 8; ++v) {
      int i = v + 8 * half;
      float qk = s[v];
      float A1 = 1.0f - 2.0f * qk + k2j;
      float Bq = 1.0f - q2v[v];
      float num2 = A1 * A1 * q2v[v] - 2.0f * A1 * Bq * qk + Bq * Bq * k2j;
      float den = fmaxf(1.0f - 2.0f * qk + q2v[v] * k2j, EPSF);
      float dn = sqrtf(fmaxf(num2, 0.0f)) / den;
      dn = fminf(fminf(dn, CLAMPV), 1.0f - EPSF);
      s_tile[w][i * 16 + l16] = -0.25f * atanhf(dn); // 2*atanh / sqrt(64)
    }
    __syncthreads();
    // online softmax, one row per lane (half 0 only)
    if (half == 0) {
      float tmax = -3.0e38f;
#pragma unroll
      for (int j = 0; j < 16; ++j) tmax = fmaxf(tmax, s_tile[w][l16 * 16 + j]);
      float m_new = fmaxf(m_r, tmax);
      float scale = __expf(m_r - m_new);
      float rsum = 0.0f;
#pragma unroll
      for (int j = 0; j < 16; ++j) {
        float p = __expf(s_tile[w][l16 * 16 + j] - m_new);
        s_tile[w][l16 * 16 + j] = p;
        rsum += p;
      }
      l_r = l_r * scale + rsum;
      m_r = m_new;
      s_scl[w][l16] = scale;
    }
    __syncthreads();
    // rescale accumulators by per-row factor
#pragma unroll
    for (int v = 0; v < 8; ++v) {
      float sc = s_scl[w][v + 8 * half];
#pragma unroll
      for (int nt = 0; nt < 4; ++nt) acc[nt][v] *= sc;
    }
    // P tile back out of LDS in A-layout
    v2f pa[4];
#pragma unroll
    for (int ks = 0; ks < 4; ++ks)
      pa[ks] = *(const v2f*)(&s_tile[w][l16 * 16 + ks * 4 + half * 2]);
    // acc += P(16x16) @ V_tile(16x64)
#pragma unroll
    for (int nt = 0; nt < 4; ++nt) {
#pragma unroll
      for (int ks = 0; ks < 4; ++ks) {
        int kr = jb + ks * 4 + half * 2;
        v2f bf;
        bf.x = Vp[(size_t)(kr + 0) * Dd + nt * 16 + l16];
        bf.y = Vp[(size_t)(kr + 1) * Dd + nt * 16 + l16];
        acc[nt] = __builtin_amdgcn_wmma_f32_16x16x4_f32(false, pa[ks], false, bf,
                                                        (short)0, acc[nt],
                                                        false, false);
      }
    }
    __syncthreads();
  }
  if (half == 0) s_l[w][l16] = l_r;
  __syncthreads();
  float* Op = O + (size_t)(b * Tt) * Dd + h * DK;
#pragma unroll
  for (int v = 0; v < 8; ++v) {
    int i = v + 8 * half;
    float inv = 1.0f / s_l[w][i];
#pragma unroll
    for (int nt = 0; nt < 4; ++nt)
      Op[(size_t)(qt + i) * Dd + nt * 16 + l16] = acc[nt][v] * inv;
  }
}

// ---------------------------------------------------------------- launcher
extern "C" void kernel_launch(void* const* d_in, const int* in_sizes, int n_in,
                              void* d_out, int out_size, void* d_ws, size_t ws_size,
                              hipStream_t stream) {
  const float* x  = (const float*)d_in[0];
  const float* Wq = (const float*)d_in[1];
  const float* bq = (const float*)d_in[2];
  const float* Wk = (const float*)d_in[3];
  const float* bk = (const float*)d_in[4];
  const float* Wv = (const float*)d_in[5];
  const float* bv = (const float*)d_in[6];
  const float* Wo = (const float*)d_in[7];
  const float* bo = (const float*)d_in[8];
  float* out = (float*)d_out;

  float* ws = (float*)d_ws;
  const size_t MAT = (size_t)NR * Dd;       // 2M floats
  float* u    = ws;                          // logmap0(x)
  float* Qb   = u + MAT;                     // Yq -> Q (in place)
  float* Kb   = Qb + MAT;                    // Yk -> K (in place)
  float* Vb   = Kb + MAT;                    // V
  float* Ob   = Vb + MAT;                    // attention output
  float* q2   = Ob + MAT;                    // (B*H*T)
  float* k2   = q2 + (size_t)Bb * Hh * Tt;
  float* pbq  = k2 + (size_t)Bb * Hh * Tt;   // expmap0(bq)
  float* pbk  = pbq + Dd;
  float* metaq = pbk + Dd;                   // |pbq|^2
  float* metak = metaq + 1;

  bias_ball_kernel<<<1, 256, 0, stream>>>(bq, pbq, metaq);
  bias_ball_kernel<<<1, 256, 0, stream>>>(bk, pbk, metak);
  logmap_kernel<<<NR, 256, 0, stream>>>(x, u);

  dim3 gg(Dd / 32, NR / 64);
  gemm_wmma_kernel<<<gg, 256, 0, stream>>>(u, Wq, nullptr, Qb, NR, Dd, Dd);
  gemm_wmma_kernel<<<gg, 256, 0, stream>>>(u, Wk, nullptr, Kb, NR, Dd, Dd);
  gemm_wmma_kernel<<<gg, 256, 0, stream>>>(u, Wv, bv, Vb, NR, Dd, Dd);

  hyp_epilogue_kernel<<<NR, 256, 0, stream>>>(Qb, pbq, metaq);
  hyp_epilogue_kernel<<<NR, 256, 0, stream>>>(Kb, pbk, metak);

  headnorm_kernel<<<NR * Hh, 64, 0, stream>>>(Qb, q2);
  headnorm_kernel<<<NR * Hh, 64, 0, stream>>>(Kb, k2);

  dim3 ga(Tt / 128, Bb * Hh);
  attn_kernel<<<ga, 256, 0, stream>>>(Qb, Kb, Vb, q2, k2, Ob);

  gemm_wmma_kernel<<<gg, 256, 0, stream>>>(Ob, Wo, bo, out, NR, Dd, Dd);
}